// GatedTestLayer_51402168598956
// MI455X (gfx1250) — compile-verified
//
#include <hip/hip_runtime.h>
#include <math.h>

#define NN 100000
#define NE 1600000
#define DIM 64
#define LPAD 68   // padded LDS row stride (floats): banks spread, no cross-half conflicts

typedef __attribute__((ext_vector_type(2))) float v2f;
typedef __attribute__((ext_vector_type(8))) float v8f;
typedef __attribute__((ext_vector_type(4))) int   v4i;

// ---------------------------------------------------------------------------
// Async global->LDS copy (GLOBAL_LOAD_ASYNC_TO_LDS_B128, ASYNCcnt-tracked).
// Probe result: builtin takes (v4i addrspace(1)*, v4i addrspace(3)*, Ii, Ii).
// ---------------------------------------------------------------------------
#if __has_builtin(__builtin_amdgcn_global_load_async_to_lds_b128)
#define USE_ASYNC_LDS 1
typedef __attribute__((address_space(1))) v4i as1_v4i;
typedef __attribute__((address_space(3))) v4i as3_v4i;
#else
#define USE_ASYNC_LDS 0
#endif

__device__ __forceinline__ void copy_b128_to_lds(void* lds_dst, const void* gsrc)
{
#if USE_ASYNC_LDS
  __builtin_amdgcn_global_load_async_to_lds_b128(
      (as1_v4i*)gsrc, (as3_v4i*)lds_dst, 0, 0);
#else
  *(float4*)lds_dst = *(const float4*)gsrc;
#endif
}

__device__ __forceinline__ void wait_async0()
{
#if USE_ASYNC_LDS
#if __has_builtin(__builtin_amdgcn_s_wait_asynccnt)
  __builtin_amdgcn_s_wait_asynccnt(0);
#else
  asm volatile("s_wait_asynccnt 0" ::: "memory");
#endif
#endif
}

__device__ __forceinline__ float sigmoidf_(float x) { return 1.0f / (1.0f + __expf(-x)); }

// ---------------------------------------------------------------------------
// Shared WMMA core: one wave computes a 16x64 output tile of (X[16x64] @ W[64x64])
// using V_WMMA_F32_16X16X4_F32. A-operand: 16x4 fp32 (2 VGPRs/lane),
// B-operand: 4x16 fp32 (2 VGPRs/lane), C/D: 16x16 fp32 (8 VGPRs/lane).
// Layouts per CDNA5 ISA §7.12.2.
// ---------------------------------------------------------------------------
__device__ __forceinline__ void wmma_tile_16x64(
    const float* __restrict__ xrow,   // &lds_x[(tile*16 + mrow) * LPAD]
    const float* __restrict__ wl,     // lds weights [64][LPAD]
    int half, int mrow, v8f acc[4])
{
#pragma unroll
  for (int k0 = 0; k0 < DIM; k0 += 4) {
    v2f a;
    a[0] = xrow[k0 + 2 * half];        // A VGPR0: K = k0 (+2 upper half)
    a[1] = xrow[k0 + 2 * half + 1];    // A VGPR1: K = k0+1 (+2 upper half)
    const float* w0 = &wl[(k0 + 2 * half) * LPAD];
    const float* w1 = &wl[(k0 + 2 * half + 1) * LPAD];
#pragma unroll
    for (int nt = 0; nt < 4; ++nt) {
      v2f b;
      b[0] = w0[nt * 16 + mrow];
      b[1] = w1[nt * 16 + mrow];
      acc[nt] = __builtin_amdgcn_wmma_f32_16x16x4_f32(
          false, a, false, b, (short)0, acc[nt], false, false);
    }
  }
}

// ---------------------------------------------------------------------------
// Kernel 1: node GEMM  out[N,64] = h @ W + b   (launched 4x: A, B, D, E)
// 256 threads = 8 waves; block covers 128 rows; each wave one 16-row tile.
// Tiles are staged into LDS with async b128 copies.
// ---------------------------------------------------------------------------
__global__ __launch_bounds__(256)
void node_gemm_kernel(const float* __restrict__ h,
                      const float* __restrict__ W,
                      const float* __restrict__ bias,
                      float* __restrict__ out, int nrows)
{
  __shared__ float wl[DIM * LPAD];
  __shared__ float hl[128 * LPAD];

  const int tid = threadIdx.x;
  const int lane = tid & 31;
  const int wave = tid >> 5;
  const int half = lane >> 4;
  const int mrow = lane & 15;
  const int blockRow = blockIdx.x * 128;

  // weights: 64 rows x 16 b128 chunks
  for (int i = tid; i < DIM * 16; i += 256) {
    int k = i >> 4, c = (i & 15) * 4;
    copy_b128_to_lds(&wl[k * LPAD + c], &W[k * DIM + c]);
  }
  // h tile: 128 rows x 16 b128 chunks (tail rows skipped; never read since NN%16==0)
  for (int i = tid; i < 128 * 16; i += 256) {
    int r = i >> 4, c = (i & 15) * 4;
    int gr = blockRow + r;
    if (gr < nrows)
      copy_b128_to_lds(&hl[r * LPAD + c], &h[(size_t)gr * DIM + c]);
  }
  wait_async0();
  __syncthreads();

  const int tileRow = blockRow + wave * 16;   // wave-uniform -> full EXEC in WMMA
  if (tileRow < nrows) {
    v8f acc[4] = {};
    wmma_tile_16x64(&hl[(wave * 16 + mrow) * LPAD], wl, half, mrow, acc);
#pragma unroll
    for (int nt = 0; nt < 4; ++nt) {
      const int col = nt * 16 + mrow;
      const float bv = bias[col];
#pragma unroll
      for (int r = 0; r < 8; ++r) {
        const int row = tileRow + r + 8 * half;   // C/D layout: M = r (+8 upper half)
        out[(size_t)row * DIM + col] = acc[nt][r] + bv;
      }
    }
  }
}

// ---------------------------------------------------------------------------
// Kernel 2: fused edge pass.
//   Ce = e @ Cw + Cb (WMMA);  e_new = Ce + Dh[src] + Eh[dst]  (NT-stored)
//   sigma = sigmoid(e_new) -> atomicAdd sum_sigma[dst]
//   BN(e) column sum / sumsq via LDS atomics, flushed with global atomics.
// ---------------------------------------------------------------------------
__global__ __launch_bounds__(256)
void edge_gemm_kernel(const float* __restrict__ e,
                      const int* __restrict__ src, const int* __restrict__ dst,
                      const float* __restrict__ Cw, const float* __restrict__ Cb,
                      const float* __restrict__ Dh, const float* __restrict__ Eh,
                      float* __restrict__ e_new, float* __restrict__ sum_sigma,
                      float* __restrict__ e_stats, int nedges)
{
  __shared__ float wl[DIM * LPAD];
  __shared__ float el[128 * LPAD];
  __shared__ int sArr[128];
  __shared__ int dArr[128];
  __shared__ float bsum[DIM];
  __shared__ float bsq[DIM];

  const int tid = threadIdx.x;
  const int lane = tid & 31;
  const int wave = tid >> 5;
  const int half = lane >> 4;
  const int mrow = lane & 15;
  const int blockRow = blockIdx.x * 128;

  if (tid < DIM) { bsum[tid] = 0.0f; bsq[tid] = 0.0f; }
  for (int i = tid; i < DIM * 16; i += 256) {
    int k = i >> 4, c = (i & 15) * 4;
    copy_b128_to_lds(&wl[k * LPAD + c], &Cw[k * DIM + c]);
  }
  for (int i = tid; i < 128 * 16; i += 256) {
    int r = i >> 4, c = (i & 15) * 4;
    int ge = blockRow + r;
    if (ge < nedges)
      copy_b128_to_lds(&el[r * LPAD + c], &e[(size_t)ge * DIM + c]);
  }
  for (int i = tid; i < 128; i += 256) {
    int ge = blockRow + i;
    sArr[i] = (ge < nedges) ? src[ge] : 0;
    dArr[i] = (ge < nedges) ? dst[ge] : 0;
  }
  wait_async0();
  __syncthreads();

  const int tileRow = blockRow + wave * 16;
  if (tileRow < nedges) {
    v8f acc[4] = {};
    wmma_tile_16x64(&el[(wave * 16 + mrow) * LPAD], wl, half, mrow, acc);
#pragma unroll
    for (int nt = 0; nt < 4; ++nt) {
      const int col = nt * 16 + mrow;
      const float bv = Cb[col];
#pragma unroll
      for (int r = 0; r < 8; ++r) {
        const int M = r + 8 * half;
        const int edge = tileRow + M;
        const int s = sArr[wave * 16 + M];
        const int d = dArr[wave * 16 + M];
        // Dh/Eh gathers: node tensors are L2-resident (25.6 MB each, 192 MB L2)
        float val = acc[nt][r] + bv + Dh[(size_t)s * DIM + col]
                                    + Eh[(size_t)d * DIM + col];
        // streaming 410 MB tensor: non-temporal, keep L2 for node tensors
        __builtin_nontemporal_store(val, &e_new[(size_t)edge * DIM + col]);
        atomicAdd(&sum_sigma[(size_t)d * DIM + col], sigmoidf_(val));
        atomicAdd(&bsum[col], val);
        atomicAdd(&bsq[col], val * val);
      }
    }
  }
  __syncthreads();
  if (tid < DIM) {
    atomicAdd(&e_stats[tid], bsum[tid]);
    atomicAdd(&e_stats[DIM + tid], bsq[tid]);
  }
}

// ---------------------------------------------------------------------------
// Kernel 3: gated message + planar reducer accumulation into fsum[dst].
// ---------------------------------------------------------------------------
__global__ __launch_bounds__(256)
void edge_message_kernel(const float* __restrict__ e_new,
                         const int* __restrict__ src, const int* __restrict__ dst,
                         const float* __restrict__ Bh, const float* __restrict__ sum_sigma,
                         const float* __restrict__ pw, const float* __restrict__ pb,
                         float* __restrict__ fsum, size_t total)
{
  size_t idx = (size_t)blockIdx.x * 256 + threadIdx.x;
  if (idx >= total) return;
  const int col = (int)(idx & 63);
  const size_t edge = idx >> 6;
  const float sg = sigmoidf_(__builtin_nontemporal_load(&e_new[idx]));
  const int s = src[edge];
  const int d = dst[edge];
  const float eee = Bh[(size_t)s * DIM + col] /
                    (sum_sigma[(size_t)s * DIM + col] + 1e-6f);
  const float m = eee * sg;
  const float w = __expf(pw[col]);
  const float f = sigmoidf_(w * fabsf(m) + pb[0]);
  atomicAdd(&fsum[(size_t)d * DIM + col], f);
}

// ---------------------------------------------------------------------------
// Kernel 4: node reducer finalize: out_h from fsum, h_new = Ah + out_h,
//           fused BN(h) column stats via LDS reduce + global atomics.
// ---------------------------------------------------------------------------
__global__ __launch_bounds__(256)
void node_finalize_kernel(const float* __restrict__ Ah, const float* __restrict__ fsum,
                          const float* __restrict__ pw, const float* __restrict__ pb,
                          float* __restrict__ h_new, float* __restrict__ h_stats,
                          size_t total)
{
  __shared__ float bsum[DIM];
  __shared__ float bsq[DIM];
  const int tid = threadIdx.x;
  if (tid < DIM) { bsum[tid] = 0.0f; bsq[tid] = 0.0f; }
  __syncthreads();
  size_t idx = (size_t)blockIdx.x * 256 + tid;
  if (idx < total) {
    const int col = (int)(idx & 63);
    float fs = fminf(fmaxf(fsum[idx], 0.001f), 0.999f);
    const float w = __expf(pw[col]);
    const float oh = (__logf(fs / (1.0f - fs)) - pb[0]) / w;
    const float hn = Ah[idx] + oh;
    h_new[idx] = hn;
    atomicAdd(&bsum[col], hn);
    atomicAdd(&bsq[col], hn * hn);
  }
  __syncthreads();
  if (tid < DIM) {
    atomicAdd(&h_stats[tid], bsum[tid]);
    atomicAdd(&h_stats[DIM + tid], bsq[tid]);
  }
}

// ---------------------------------------------------------------------------
// Kernel 5: finalize BN statistics (mean, rsqrt(var+eps)) for h and e.
// stats layout: [0:64)=hsum [64:128)=hsumsq [128:192)=esum [192:256)=esumsq
//               [256:320)=mu_h [320:384)=inv_h [384:448)=mu_e [448:512)=inv_e
// ---------------------------------------------------------------------------
__global__ void stats_finalize_kernel(float* __restrict__ stats)
{
  const int c = threadIdx.x;  // 64 threads
  const float muh = stats[c] / (float)NN;
  const float varh = stats[64 + c] / (float)NN - muh * muh;
  stats[256 + c] = muh;
  stats[320 + c] = rsqrtf(varh + 1e-5f);
  const float mue = stats[128 + c] / (float)NE;
  const float vare = stats[192 + c] / (float)NE - mue * mue;
  stats[384 + c] = mue;
  stats[448 + c] = rsqrtf(vare + 1e-5f);
}

// ---------------------------------------------------------------------------
// Kernels 6/7: BN apply + ReLU + residual (streaming: NT loads/stores).
// ---------------------------------------------------------------------------
__global__ __launch_bounds__(256)
void h_out_kernel(const float* __restrict__ h, const float* __restrict__ h_new,
                  const float* __restrict__ stats,
                  const float* __restrict__ gamma, const float* __restrict__ beta,
                  float* __restrict__ out, size_t total)
{
  size_t idx = (size_t)blockIdx.x * 256 + threadIdx.x;
  if (idx >= total) return;
  const int col = (int)(idx & 63);
  float y = (h_new[idx] - stats[256 + col]) * stats[320 + col] * gamma[col] + beta[col];
  __builtin_nontemporal_store(h[idx] + fmaxf(y, 0.0f), &out[idx]);
}

__global__ __launch_bounds__(256)
void e_out_kernel(const float* __restrict__ e, const float* __restrict__ e_new,
                  const float* __restrict__ stats,
                  const float* __restrict__ gamma, const float* __restrict__ beta,
                  float* __restrict__ out, size_t total)
{
  size_t idx = (size_t)blockIdx.x * 256 + threadIdx.x;
  if (idx >= total) return;
  const int col = (int)(idx & 63);
  const float en = __builtin_nontemporal_load(&e_new[idx]);
  const float ev = __builtin_nontemporal_load(&e[idx]);
  float y = (en - stats[384 + col]) * stats[448 + col] * gamma[col] + beta[col];
  __builtin_nontemporal_store(ev + fmaxf(y, 0.0f), &out[idx]);
}

// ---------------------------------------------------------------------------
extern "C" void kernel_launch(void* const* d_in, const int* in_sizes, int n_in,
                              void* d_out, int out_size, void* d_ws, size_t ws_size,
                              hipStream_t stream)
{
  const float* h  = (const float*)d_in[0];
  const float* e  = (const float*)d_in[1];
  const int* src  = (const int*)d_in[2];
  const int* dst  = (const int*)d_in[3];
  const float* Aw = (const float*)d_in[4];  const float* Ab = (const float*)d_in[5];
  const float* Bw = (const float*)d_in[6];  const float* Bb = (const float*)d_in[7];
  const float* Cw = (const float*)d_in[8];  const float* Cb = (const float*)d_in[9];
  const float* Dw = (const float*)d_in[10]; const float* Db = (const float*)d_in[11];
  const float* Ew = (const float*)d_in[12]; const float* Eb = (const float*)d_in[13];
  const float* pw = (const float*)d_in[14]; const float* pb = (const float*)d_in[15];
  const float* bn_h_g = (const float*)d_in[16]; const float* bn_h_b = (const float*)d_in[17];
  const float* bn_e_g = (const float*)d_in[18]; const float* bn_e_b = (const float*)d_in[19];

  const size_t NDf = (size_t)NN * DIM;   // 6,400,000
  const size_t EDf = (size_t)NE * DIM;   // 102,400,000

  float* ws        = (float*)d_ws;
  float* Ah        = ws;
  float* Bh        = ws + 1 * NDf;
  float* Dh        = ws + 2 * NDf;
  float* Eh        = ws + 3 * NDf;
  float* h_new     = ws + 4 * NDf;
  float* sum_sigma = ws + 5 * NDf;
  float* fsum      = ws + 6 * NDf;
  float* stats     = ws + 7 * NDf;           // 512 floats
  float* e_new     = ws + 7 * NDf + 1024;

  // Zero the accumulators every launch (graph-capture-safe memset nodes).
  (void)hipMemsetAsync(sum_sigma, 0, 2 * NDf * sizeof(float), stream);  // sum_sigma + fsum
  (void)hipMemsetAsync(stats, 0, 256 * sizeof(float), stream);

  const dim3 blk(256);
  const int ngrid = (NN + 127) / 128;
  node_gemm_kernel<<<ngrid, blk, 0, stream>>>(h, Aw, Ab, Ah, NN);
  node_gemm_kernel<<<ngrid, blk, 0, stream>>>(h, Bw, Bb, Bh, NN);
  node_gemm_kernel<<<ngrid, blk, 0, stream>>>(h, Dw, Db, Dh, NN);
  node_gemm_kernel<<<ngrid, blk, 0, stream>>>(h, Ew, Eb, Eh, NN);

  const int egrid = (NE + 127) / 128;
  edge_gemm_kernel<<<egrid, blk, 0, stream>>>(e, src, dst, Cw, Cb, Dh, Eh,
                                              e_new, sum_sigma, stats + 128, NE);

  edge_message_kernel<<<(unsigned)((EDf + 255) / 256), blk, 0, stream>>>(
      e_new, src, dst, Bh, sum_sigma, pw, pb, fsum, EDf);

  node_finalize_kernel<<<(unsigned)((NDf + 255) / 256), blk, 0, stream>>>(
      Ah, fsum, pw, pb, h_new, stats, NDf);

  stats_finalize_kernel<<<1, 64, 0, stream>>>(stats);

  h_out_kernel<<<(unsigned)((NDf + 255) / 256), blk, 0, stream>>>(
      h, h_new, stats, bn_h_g, bn_h_b, (float*)d_out, NDf);
  e_out_kernel<<<(unsigned)((EDf + 255) / 256), blk, 0, stream>>>(
      e, e_new, stats, bn_e_g, bn_e_b, (float*)d_out + NDf, EDf);
}